// MinibatchDiscrimination_970662608991
// MI455X (gfx1250) — compile-verified
//
#include <hip/hip_runtime.h>
#include <hip/hip_bf16.h>

typedef __attribute__((ext_vector_type(2))) float v2f;
typedef __attribute__((ext_vector_type(8))) float v8f;

#define BATCH 256
#define IN_F  1024
#define OUT_F 64
#define KD    16
#define OUT_ROW (IN_F + OUT_F)   // 1088

// ---------------------------------------------------------------------------
// Kernel 1: M[256,1024] = x[256,1024] @ W[1024,1024]  (W = T viewed row-major)
// One wave per 16x16 output tile, K-loop of 256 x V_WMMA_F32_16X16X4_F32.
// ---------------------------------------------------------------------------
__global__ __launch_bounds__(256) void mbd_gemm_wmma(
    const float* __restrict__ x,   // [256,1024]
    const float* __restrict__ W,   // [1024,1024] row-major (col = o*16 + k)
    float* __restrict__ M)         // [256,1024]
{
    const int lane = threadIdx.x & 31;
    const int wave = blockIdx.x * (blockDim.x >> 5) + (threadIdx.x >> 5);
    const int tm   = wave >> 6;    // 0..15  (row tile)
    const int tn   = wave & 63;    // 0..63  (col tile)

    const int half = lane >> 4;    // 0 or 1
    const int l16  = lane & 15;
    const int kb   = half * 2;     // K pair base per ISA A/B layout

    const int rowA = tm * 16 + l16;
    const int colB = tn * 16 + l16;

    const float* aptr = x + rowA * IN_F + kb;       // contiguous K pair
    const float* bptr = W + kb * IN_F + colB;       // strided K pair

    v8f c = {};
    #pragma unroll 4
    for (int ki = 0; ki < IN_F / 4; ++ki) {
        // A: x[rowA][ki*4 + kb], x[rowA][ki*4 + kb + 1]  (8B aligned float2)
        v2f a = *(const v2f*)(aptr + ki * 4);
        // B: W[ki*4 + kb][colB], W[ki*4 + kb + 1][colB]
        v2f b;
        b.x = bptr[ki * 4 * IN_F];
        b.y = bptr[ki * 4 * IN_F + IN_F];
        c = __builtin_amdgcn_wmma_f32_16x16x4_f32(
                /*neg_a=*/false, a, /*neg_b=*/false, b,
                /*c_mod=*/(short)0, c, /*reuse_a=*/false, /*reuse_b=*/false);
    }

    // C/D layout: VGPR v holds row (v + 8*half), col = l16
    float* outp = M + (tm * 16 + half * 8) * IN_F + tn * 16 + l16;
    #pragma unroll
    for (int v = 0; v < 8; ++v)
        outp[v * IN_F] = c[v];
}

// ---------------------------------------------------------------------------
// Kernel 2: per-row pairwise L1 -> exp -> masked sum, plus copy of x row.
// Block i: 256 threads = 64 'o' columns x 4 j-stride groups.
// ---------------------------------------------------------------------------
__global__ __launch_bounds__(256) void mbd_pairwise(
    const float* __restrict__ x,   // [256,1024]
    const float* __restrict__ M,   // [256,1024]
    float* __restrict__ out)       // [256,1088]
{
    __shared__ float mi[IN_F];         // M[i,:]
    __shared__ float red[4][OUT_F];    // partial sums over j-groups

    const int i = blockIdx.x;
    const int t = threadIdx.x;

    // Copy x row into out[:, 0:1024] and stage M[i,:] in LDS (float4 each).
    ((float4*)(out + i * OUT_ROW))[t] = ((const float4*)(x + i * IN_F))[t];
    ((float4*)mi)[t] = ((const float4*)(M + i * IN_F))[t];
    __syncthreads();

    const int o  = t & 63;   // output feature
    const int js = t >> 6;   // j-stride group 0..3

    float a[KD];
    #pragma unroll
    for (int k = 0; k < KD; ++k) a[k] = mi[o * KD + k];

    float acc = 0.0f;
    for (int j = js; j < BATCH; j += 4) {
        const float4* mj = (const float4*)(M + j * IN_F + o * KD);
        float L1 = 0.0f;
        #pragma unroll
        for (int q = 0; q < KD / 4; ++q) {
            float4 v = mj[q];
            L1 += fabsf(a[q * 4 + 0] - v.x);
            L1 += fabsf(a[q * 4 + 1] - v.y);
            L1 += fabsf(a[q * 4 + 2] - v.z);
            L1 += fabsf(a[q * 4 + 3] - v.w);
        }
        float e = __expf(-L1);
        acc += (j == i) ? 0.0f : e;
    }
    red[js][o] = acc;
    __syncthreads();

    if (t < OUT_F) {
        out[i * OUT_ROW + IN_F + t] =
            red[0][t] + red[1][t] + red[2][t] + red[3][t];
    }
}

extern "C" void kernel_launch(void* const* d_in, const int* in_sizes, int n_in,
                              void* d_out, int out_size, void* d_ws, size_t ws_size,
                              hipStream_t stream) {
    const float* x = (const float*)d_in[0];   // [256,1024]
    const float* T = (const float*)d_in[1];   // [1024,64,16] == W[1024,1024]
    float* out = (float*)d_out;               // [256,1088]
    float* M   = (float*)d_ws;                // [256,1024] scratch (1 MB)

    // GEMM: 1024 tiles -> 1024 waves -> 128 blocks of 8 waves.
    mbd_gemm_wmma<<<128, 256, 0, stream>>>(x, T, M);
    // Pairwise: one block per batch row.
    mbd_pairwise<<<BATCH, 256, 0, stream>>>(x, M, out);
}